// Model_88957362634943
// MI455X (gfx1250) — compile-verified
//
#include <hip/hip_runtime.h>
#include <math.h>

// LBS forward restructured as X(V x 92) @ B(92 x 48) GEMM via V_WMMA_F32_16X16X4_F32.
//   X[v, j*4+k] = w[v,j] * [x,y,z,1][k]
//   B[j*4+k, p*3+i] = A[p,j,i,k]   (A = FK-derived 3x4 joint transforms)
//   out[p,v,i] = (X@B)[v, p*3+i] + shift[p,i]

typedef __attribute__((ext_vector_type(2))) float v2f;
typedef __attribute__((ext_vector_type(8))) float v8f;

#define NUM_J  23
#define NPOSE  16
#define NCOL   48            // NPOSE * 3 output columns
#define KDIM   92            // NUM_J * 4
#define VTOT   500000
#define VTILE  16
#define NTILES (VTOT / VTILE)        // 31250 exactly
#define WPB    8                     // waves per block (256 threads, wave32)
#define OUT_JOFF (NPOSE * VTOT * 3)  // 24,000,000 : offset of posed-joints output

struct PoseArgs { const float* p[11]; };

// ---------------- Kernel 1: pose -> Rodrigues -> FK chain -> B matrix ----------------
__global__ void lbs_prep(const float* __restrict__ joints,
                         const float* __restrict__ disp,
                         const float* __restrict__ rnd,
                         PoseArgs pa,
                         float* __restrict__ Bws,      // [NCOL][KDIM] col-major-by-col
                         float* __restrict__ shiftws,  // [NCOL]
                         float* __restrict__ outJ)     // d_out + OUT_JOFF : (P,23,3)
{
    const int parents[NUM_J] = {-1,0,1,1,3,4,5,4,7,4,9,1,11,12,13,12,15,12,17,0,19,0,21};
    const int slotIdx[11]    = {0,3,4,5,7,9,11,12,13,15,17};
    const float HALFPI  = 1.57079632679489662f;
    const float QUARTPI = 0.78539816339744831f;
    const float scales[11] = {QUARTPI,HALFPI,HALFPI,QUARTPI,QUARTPI,QUARTPI,
                              HALFPI,HALFPI,QUARTPI,QUARTPI,QUARTPI};

    int p = threadIdx.x;
    if (p >= NPOSE) return;

    float pose[NUM_J][3];
    for (int j = 0; j < NUM_J; ++j)
        for (int c = 0; c < 3; ++c) pose[j][c] = 0.0f;
    for (int s = 0; s < 11; ++s)
        for (int c = 0; c < 3; ++c)
            pose[slotIdx[s]][c] = scales[s] * tanhf(pa.p[s][p*3 + c]);

    float jp[NUM_J][3];
    for (int j = 0; j < NUM_J; ++j)
        for (int c = 0; c < 3; ++c) jp[j][c] = joints[j*3 + c];

    float G[NUM_J][12];   // 3x4 rows, bottom row implicit [0 0 0 1]
    for (int j = 0; j < NUM_J; ++j) {
        // Rodrigues: R = I + sin*K + (1-cos)*K^2   (matches reference incl. 1e-16 eps)
        float rx = pose[j][0], ry = pose[j][1], rz = pose[j][2];
        float ang = sqrtf(rx*rx + ry*ry + rz*rz + 1e-16f);
        float inv = 1.0f / ang;
        float ax = rx*inv, ay = ry*inv, az = rz*inv;
        float sn = sinf(ang), cs = cosf(ang), t = 1.0f - cs;
        float R[9];
        R[0] = 1.0f - t*(ay*ay + az*az);
        R[1] = -sn*az + t*ax*ay;
        R[2] =  sn*ay + t*ax*az;
        R[3] =  sn*az + t*ax*ay;
        R[4] = 1.0f - t*(ax*ax + az*az);
        R[5] = -sn*ax + t*ay*az;
        R[6] = -sn*ay + t*ax*az;
        R[7] =  sn*ax + t*ay*az;
        R[8] = 1.0f - t*(ax*ax + ay*ay);

        float rel[3];
        int par = parents[j];
        for (int c = 0; c < 3; ++c)
            rel[c] = (j == 0) ? jp[0][c] : (jp[j][c] - jp[par][c]);

        if (j == 0) {
            for (int i = 0; i < 3; ++i) {
                for (int k = 0; k < 3; ++k) G[0][i*4+k] = R[i*3+k];
                G[0][i*4+3] = rel[i];
            }
        } else {
            for (int i = 0; i < 3; ++i) {
                float g0 = G[par][i*4+0], g1 = G[par][i*4+1];
                float g2 = G[par][i*4+2], g3 = G[par][i*4+3];
                for (int k = 0; k < 3; ++k)
                    G[j][i*4+k] = g0*R[0*3+k] + g1*R[1*3+k] + g2*R[2*3+k];
                G[j][i*4+3] = g0*rel[0] + g1*rel[1] + g2*rel[2] + g3;
            }
        }
    }

    float sh[3];
    for (int c = 0; c < 3; ++c) {
        sh[c] = rnd[p*3 + c] + 3.0f * tanhf(disp[p*3 + c]);
        shiftws[p*3 + c] = sh[c];
    }

    // posed joints output (+shift)
    for (int j = 0; j < NUM_J; ++j)
        for (int i = 0; i < 3; ++i)
            outJ[p*NUM_J*3 + j*3 + i] = G[j][i*4+3] + sh[i];

    // B matrix: columns = (p,i), contiguous K per column (even-aligned float2 pairs)
    for (int j = 0; j < NUM_J; ++j) {
        for (int i = 0; i < 3; ++i) {
            int col = p*3 + i;
            float a0 = G[j][i*4+0], a1 = G[j][i*4+1], a2 = G[j][i*4+2];
            float a3 = G[j][i*4+3] - (a0*jp[j][0] + a1*jp[j][1] + a2*jp[j][2]);
            float* dst = Bws + col*KDIM + j*4;
            dst[0] = a0; dst[1] = a1; dst[2] = a2; dst[3] = a3;
        }
    }
}

// ---------------- Kernel 2: the V x 48 x 92 fp32 WMMA GEMM ----------------
__global__ __launch_bounds__(256) void lbs_wmma(
    const float* __restrict__ verts,     // (V,3)
    const float* __restrict__ weights,   // (V,23)
    const float* __restrict__ Bws,       // [NCOL*KDIM]
    const float* __restrict__ shiftws,   // [NCOL]
    float* __restrict__ out)             // (P,V,3)
{
    __shared__ float BL[NCOL * KDIM];    // 17,664 B
    __shared__ float SL[NCOL];

    int tid = threadIdx.x;
    for (int i = tid; i < NCOL*KDIM; i += 256) BL[i] = Bws[i];
    if (tid < NCOL) SL[tid] = shiftws[tid];
    __syncthreads();

    int wave = tid >> 5;
    int lane = tid & 31;
    int half = lane >> 4;          // A/B fragment half: K {0,1} vs {2,3}
    int lm   = lane & 15;

    int tile = blockIdx.x * WPB + wave;   // wave-uniform -> EXEC stays all-1s
    if (tile >= NTILES) return;

    int v0 = tile * VTILE;
    int v  = v0 + lm;

    // Per-lane X-row ingredients: 23 weights + coordinate pair for this half
    float w[NUM_J];
    #pragma unroll
    for (int j = 0; j < NUM_J; ++j) w[j] = weights[v*NUM_J + j];
    float vx = verts[v*3+0], vy = verts[v*3+1], vz = verts[v*3+2];
    float c0 = half ? vz : vx;     // K = j*4 + half*2 + 0
    float c1 = half ? 1.0f : vy;   // K = j*4 + half*2 + 1

    #pragma unroll
    for (int nt = 0; nt < 3; ++nt) {
        v8f C = {0.0f,0.0f,0.0f,0.0f,0.0f,0.0f,0.0f,0.0f};
        int col = nt*16 + lm;
        const float* bp = &BL[col*KDIM + half*2];   // even dword -> 8B aligned

        #pragma unroll
        for (int j = 0; j < NUM_J; ++j) {           // K = 92 in 23 steps of 4
            v2f a; a.x = w[j]*c0; a.y = w[j]*c1;
            v2f b = *(const v2f*)(bp + j*4);        // ds_load_b64, conflict-free
            C = __builtin_amdgcn_wmma_f32_16x16x4_f32(
                    false, a, false, b, (short)0, C, false, false);
        }

        float sh = SL[col];                         // shift depends on column only
        int pp = col / 3;
        int ci = col - pp*3;
        long base = (long)pp * (long)(VTOT*3) + (long)v0*3 + ci;
        #pragma unroll
        for (int r = 0; r < 8; ++r) {
            int row = r + half*8;                   // C layout: VGPR r -> M=r / M=r+8
            out[base + (long)row*3] = C[r] + sh;
        }
    }
}

extern "C" void kernel_launch(void* const* d_in, const int* in_sizes, int n_in,
                              void* d_out, int out_size, void* d_ws, size_t ws_size,
                              hipStream_t stream) {
    const float* verts   = (const float*)d_in[0];
    const float* joints  = (const float*)d_in[1];
    const float* weights = (const float*)d_in[2];
    const float* disp    = (const float*)d_in[3];
    const float* rnd     = (const float*)d_in[4];
    PoseArgs pa;
    for (int s = 0; s < 11; ++s) pa.p[s] = (const float*)d_in[5 + s];

    float* Bws     = (float*)d_ws;            // NCOL*KDIM floats
    float* shiftws = Bws + NCOL*KDIM;         // NCOL floats
    float* out     = (float*)d_out;

    lbs_prep<<<1, 32, 0, stream>>>(joints, disp, rnd, pa, Bws, shiftws, out + OUT_JOFF);

    int blocks = (NTILES + WPB - 1) / WPB;    // 3907
    lbs_wmma<<<blocks, 256, 0, stream>>>(verts, weights, Bws, shiftws, out);
}